// UniformMatchingLoss_11544872092219
// MI455X (gfx1250) — compile-verified
//
#include <hip/hip_runtime.h>
#include <hip/hip_bf16.h>

// Problem size is fixed by the reference: n = 16384 float32, eps = 0.1.
#define N        16384
#define NT       1024
#define INV_EPS  10.0f   // 1 / REG_STRENGTH

// ws layout (floats):
//   ws[0]            : flag (int bits)
//   ws[4 .. 4+N)     : sorted x   (16-byte aligned for float4 spills)
//   ws[4+N .. 4+2N)  : PAV sums stack
//   ws[4+2N .. 4+3N) : PAV counts stack
#define WS_SORT 4

// Compiler-only ordering fence for intra-wave LDS communication.
// Hardware keeps a wave's LDS ops in order (ISA: DScnt, in-order); we only
// need to stop the compiler from caching/reordering across steps.
__device__ __forceinline__ void wave_sync() {
  __builtin_amdgcn_wave_barrier();
  asm volatile("" ::: "memory");
}

__device__ __forceinline__ void order2(float& a, float& b, bool up) {
  float lo = fminf(a, b);
  float hi = fmaxf(a, b);
  a = up ? lo : hi;
  b = up ? hi : lo;
}

// Register tail of a bitonic phase: j = 8,4,2,1 on this thread's 16 elements.
// Direction is constant per thread for k >= 32.
__device__ __forceinline__ void local_tail(float* r, float4* vb, int base, int k) {
  ((float4*)r)[0] = vb[0]; ((float4*)r)[1] = vb[1];
  ((float4*)r)[2] = vb[2]; ((float4*)r)[3] = vb[3];
  bool up = ((base & k) == 0);
#pragma unroll
  for (int j = 8; j >= 1; j >>= 1) {
#pragma unroll
    for (int e = 0; e < 16; ++e) {
      if (!(e & j)) order2(r[e], r[e | j], up);
    }
  }
  vb[0] = ((float4*)r)[0]; vb[1] = ((float4*)r)[1];
  vb[2] = ((float4*)r)[2]; vb[3] = ((float4*)r)[3];
}

// ---------------------------------------------------------------------------
// Kernel 1: one workgroup, whole problem LDS-resident on one WGP.
//   - per-wave async global->LDS fill (GLOBAL_LOAD_ASYNC_TO_LDS_B128)
//   - k=2..16   : registers (no LDS, no barriers)
//   - k=32..512 : wave-local LDS steps (j>=16) + register tail, NO s_barrier
//   - k=1024..N : s_barrier only for j>=512 cross-wave steps
//   - shuffle-based max reduction (1 barrier)
// ---------------------------------------------------------------------------
__global__ __launch_bounds__(NT)
void sort_loss_kernel(const float* __restrict__ x,
                      float* __restrict__ out,
                      float* __restrict__ ws) {
  __shared__ __align__(16) float v[N];
  __shared__ float wmax[NT / 32];
  __shared__ int flag;
  const int tid   = threadIdx.x;
  const int lane  = tid & 31;
  const int wid   = tid >> 5;
  const int base  = tid << 4;   // this thread's 16-element block
  const int rbase = wid << 9;   // this wave's 512-element region
  if (tid == 0) flag = 0;

  // ---- per-wave async global -> LDS fill of own block (4 x b128/lane) ----
#pragma unroll
  for (int c = 0; c < 4; ++c) {
    unsigned lds_off = (unsigned)(uintptr_t)&v[base + 4 * c];  // LDS offset
    unsigned long long ga = (unsigned long long)(uintptr_t)&x[base + 4 * c];
    asm volatile("global_load_async_to_lds_b128 %0, %1, off"
                 :: "v"(lds_off), "v"(ga) : "memory");
  }
  asm volatile("s_wait_asynccnt 0x0" ::: "memory");  // ASYNCcnt is per-wave
  wave_sync();

  float r[16];
  float4* vb = (float4*)(v + base);

  // ---- Phase A: k = 2..16 fully in registers ----
  ((float4*)r)[0] = vb[0]; ((float4*)r)[1] = vb[1];
  ((float4*)r)[2] = vb[2]; ((float4*)r)[3] = vb[3];
#pragma unroll
  for (int k = 2; k <= 16; k <<= 1) {
#pragma unroll
    for (int j = k >> 1; j >= 1; j >>= 1) {
#pragma unroll
      for (int e = 0; e < 16; ++e) {
        if (!(e & j)) order2(r[e], r[e | j], (((base + e) & k) == 0));
      }
    }
  }
  vb[0] = ((float4*)r)[0]; vb[1] = ((float4*)r)[1];
  vb[2] = ((float4*)r)[2]; vb[3] = ((float4*)r)[3];
  wave_sync();

  // ---- Phase B: k = 32..512, fully wave-local (no s_barrier) ----
  for (int k = 32; k <= 512; k <<= 1) {
    for (int j = k >> 1; j >= 16; j >>= 1) {
      for (int q = lane; q < 256; q += 32) {          // 256 pairs per region
        int i = rbase + (((q & ~(j - 1)) << 1) | (q & (j - 1)));
        int m = i | j;
        float a = v[i];
        float b = v[m];
        bool up = ((i & k) == 0);
        if (up ? (a > b) : (a < b)) { v[i] = b; v[m] = a; }
      }
      wave_sync();
    }
    local_tail(r, vb, base, k);
    wave_sync();
  }
  __syncthreads();   // first workgroup barrier: regions now interact

  // ---- Phase C: k = 1024..N ----
  for (int k = 1024; k <= N; k <<= 1) {
    // cross-wave steps
    for (int j = k >> 1; j >= 512; j >>= 1) {
      for (int p = tid; p < N / 2; p += NT) {
        int i = ((p & ~(j - 1)) << 1) | (p & (j - 1));
        int m = i | j;
        float a = v[i];
        float b = v[m];
        bool up = ((i & k) == 0);
        if (up ? (a > b) : (a < b)) { v[i] = b; v[m] = a; }
      }
      __syncthreads();
    }
    // wave-local steps j = 256..16
    for (int j = 256; j >= 16; j >>= 1) {
      for (int q = lane; q < 256; q += 32) {
        int i = rbase + (((q & ~(j - 1)) << 1) | (q & (j - 1)));
        int m = i | j;
        float a = v[i];
        float b = v[m];
        bool up = ((i & k) == 0);
        if (up ? (a > b) : (a < b)) { v[i] = b; v[m] = a; }
      }
      wave_sync();
    }
    local_tail(r, vb, base, k);
    __syncthreads();   // phase boundary (also publishes final sorted array)
  }
  // r holds this thread's fully-sorted block [base, base+16)

  // ---- loss + PAV-degeneracy check + aligned spill for fallback ----
  float lmax = 0.0f;
  int viol = 0;
  const float inv_n = 1.0f / (float)N;
#pragma unroll
  for (int e = 0; e < 16; ++e) {
    int i = base + e;
    lmax = fmaxf(lmax, fabsf((float)(i + 1) * inv_n - r[e]));
    if (i + 1 < N) {
      float nxt = (e < 15) ? r[e + 1] : v[i + 1];   // neighbor block head
      if ((nxt - r[e]) >= INV_EPS) viol = 1;        // PAV would merge
    }
  }
  {
    float4* wb = (float4*)(ws + WS_SORT + base);
    wb[0] = ((float4*)r)[0]; wb[1] = ((float4*)r)[1];
    wb[2] = ((float4*)r)[2]; wb[3] = ((float4*)r)[3];
  }

  // ---- wave32 shuffle reduction, then one cross-wave step ----
#pragma unroll
  for (int o = 16; o >= 1; o >>= 1)
    lmax = fmaxf(lmax, __shfl_xor(lmax, o, 32));
  unsigned long long wv = __ballot(viol);
  if (lane == 0) {
    wmax[wid] = lmax;
    if (wv) atomicOr(&flag, 1);
  }
  __syncthreads();
  if (wid == 0) {
    float m = wmax[lane];                 // NT/32 == 32 partial maxima
#pragma unroll
    for (int o = 16; o >= 1; o >>= 1)
      m = fmaxf(m, __shfl_xor(m, o, 32));
    if (lane == 0) {
      ws[0] = __int_as_float(flag);
      if (flag == 0) out[0] = m;          // degenerate (expected) path: done
    }
  }
}

// ---------------------------------------------------------------------------
// Kernel 2: serial PAV fallback. Only taken if some adjacent sorted gap
// >= 1/eps (impossible for inputs in [0,1) with eps=0.1).
// ---------------------------------------------------------------------------
__global__ void pav_fallback_kernel(float* __restrict__ out,
                                    float* __restrict__ ws) {
  if (threadIdx.x != 0 || blockIdx.x != 0) return;
  if (__float_as_int(ws[0]) == 0) return;   // fast path already wrote out

  const float* a = ws + WS_SORT;
  float* sums = ws + WS_SORT + N;
  float* cnts = ws + WS_SORT + 2 * N;

  // PAV on y[i] = -a[i] - (N-i)*INV_EPS (non-decreasing fit)
  int ptr = 0;
  for (int i = 0; i < N; ++i) {
    float y = -a[i] - (float)(N - i) * INV_EPS;
    sums[ptr] = y;
    cnts[ptr] = 1.0f;
    ++ptr;
    while (ptr > 1 &&
           (sums[ptr - 2] / cnts[ptr - 2]) >= (sums[ptr - 1] / cnts[ptr - 1])) {
      sums[ptr - 2] += sums[ptr - 1];
      cnts[ptr - 2] += cnts[ptr - 1];
      --ptr;
    }
  }

  // expand block means; out_soft[i] = -w[i] - iso(y)[i]; loss = max | . |
  float loss = 0.0f;
  const float inv_n = 1.0f / (float)N;
  int i = 0;
  for (int b = 0; b < ptr; ++b) {
    float m = sums[b] / cnts[b];
    int c = (int)(cnts[b] + 0.5f);
    for (int e = 0; e < c && i < N; ++e, ++i) {
      float o = -((float)(N - i) * INV_EPS) - m;
      loss = fmaxf(loss, fabsf((float)(i + 1) * inv_n - o));
    }
  }
  out[0] = loss;
}

extern "C" void kernel_launch(void* const* d_in, const int* in_sizes, int n_in,
                              void* d_out, int out_size, void* d_ws, size_t ws_size,
                              hipStream_t stream) {
  const float* x = (const float*)d_in[0];
  float* out = (float*)d_out;
  float* ws = (float*)d_ws;
  (void)in_sizes; (void)n_in; (void)out_size; (void)ws_size;

  sort_loss_kernel<<<1, NT, 0, stream>>>(x, out, ws);
  pav_fallback_kernel<<<1, 32, 0, stream>>>(out, ws);
}